// Yolo3_61460982006078
// MI455X (gfx1250) — compile-verified
//
#include <hip/hip_runtime.h>
#include <hip/hip_bf16.h>
#include <stdint.h>

#define SDIM 52
#define NBOX (SDIM*SDIM*3)      // 8112
#define NCH  255
#define NCLS 80
#define STRIDE_F 8.0f           // 416/52
#define CNF_T 0.5f
#define IOU_T 0.4f
#define CLS_OFF 10000.0f
#define TILE 16
#define NTILE (NBOX/TILE)       // 507 (exact: 507*16 = 8112)
#define MASKW 512               // padded 16-col words per row
#define MSORT 8192

typedef float v2f __attribute__((ext_vector_type(2)));
typedef float v8f __attribute__((ext_vector_type(8)));

// ---- workspace layout (bytes) ----
#define WS_BOXES   0u           // float4[NBOX]          129792
#define WS_AREA    129792u      // float [NBOX]           32448
#define WS_KEY     162240u      // u64   [NBOX]           64896
#define WS_ORDER   227136u      // u32   [NBOX]           32448
#define WS_SBOX    259584u      // float4[NBOX] sorted   129792
#define WS_SAREA   389376u      // float [NBOX] sorted    32448
#define WS_VCNT    421824u      // u32                        4
#define WS_MASK    421888u      // u16   [NBOX*MASKW]   8306688

// ------------------------------------------------------------------
// Kernel 1: decode.  One block per h-row; async-stage feat[:,h,:] into LDS.
// ------------------------------------------------------------------
__global__ __launch_bounds__(256)
void k_decode(const float* __restrict__ feat, const float* __restrict__ anchors,
              float* __restrict__ out, float4* __restrict__ wsBox,
              float* __restrict__ wsArea, unsigned long long* __restrict__ wsKey)
{
    __shared__ float sfeat[NCH * SDIM];          // 53040 B
    const int h   = blockIdx.x;
    const int tid = threadIdx.x;

    // feat element (c,h,w) at c*2704 + h*52 + w; each (c,h) segment = 13 x float4, 16B aligned.
    const unsigned ldsBase = (unsigned)(uintptr_t)sfeat;
    for (int t = tid; t < NCH * 13; t += 256) {
        int c = t / 13, seg = t % 13;
        unsigned long long ga =
            (unsigned long long)(uintptr_t)(feat + c * (SDIM * SDIM) + h * SDIM + seg * 4);
        unsigned la = ldsBase + (unsigned)((c * SDIM + seg * 4) * 4);
        asm volatile("global_load_async_to_lds_b128 %0, %1, off"
                     :: "v"(la), "v"(ga) : "memory");
    }
    asm volatile("s_wait_asynccnt 0" ::: "memory");
    __syncthreads();

    if (tid < SDIM * 3) {
        int w = tid / 3, a = tid % 3;
        int n = w * (SDIM * 3) + h * 3 + a;          // row index of x
        const float* p = &sfeat[a * 85 * SDIM + w];  // p[f*SDIM] = feature f
        float tx = p[0], ty = p[1*SDIM], tw = p[2*SDIM], th = p[3*SDIM], tc = p[4*SDIM];
        // reference: gx[n] = h (2nd spatial idx), gy[n] = w
        float cx = 1.f/(1.f + __expf(-tx)) + (float)h;
        float cy = 1.f/(1.f + __expf(-ty)) + (float)w;
        float bw = anchors[a*2+0] * __expf(tw);
        float bh = anchors[a*2+1] * __expf(th);
        float conf = 1.f/(1.f + __expf(-tc));
        float best = p[5*SDIM]; int bid = 0;         // first-occurrence argmax
        for (int k = 1; k < NCLS; ++k) {
            float v = p[(5+k)*SDIM];
            if (v > best) { best = v; bid = k; }
        }
        float x1 = cx - bw*0.5f, y1 = cy - bh*0.5f;
        float x2 = cx + bw*0.5f, y2 = cy + bh*0.5f;
        float* o = out + n * 7;
        o[0]=STRIDE_F*x1; o[1]=STRIDE_F*y1; o[2]=STRIDE_F*x2; o[3]=STRIDE_F*y2;
        o[4]=STRIDE_F*conf; o[5]=STRIDE_F*best; o[6]=STRIDE_F*(float)bid;

        float off = CLS_OFF * (float)bid;
        wsBox[n]  = make_float4(x1+off, y1+off, x2+off, y2+off);
        wsArea[n] = bw * bh;
        bool valid = conf > CNF_T;
        float score = valid ? best : -__builtin_inff();
        unsigned sb = __float_as_uint(score);
        unsigned su = (sb & 0x80000000u) ? ~sb : (sb | 0x80000000u); // ascending map
        // ascending key = score desc, index asc (stable argsort(-score))
        wsKey[n] = ((unsigned long long)(~su) << 32) | (unsigned)n;
    }
}

// ------------------------------------------------------------------
// Kernel 2: single-block bitonic sort of keys; gather sorted boxes; find V.
// ------------------------------------------------------------------
__global__ __launch_bounds__(1024)
void k_sort(const unsigned long long* __restrict__ wsKey,
            const float4* __restrict__ wsBox, const float* __restrict__ wsArea,
            unsigned* __restrict__ wsOrder, float4* __restrict__ sBox,
            float* __restrict__ sArea, unsigned* __restrict__ wsV)
{
    __shared__ unsigned long long sk[MSORT];       // 64 KB
    const int tid = threadIdx.x;
    for (int p = tid; p < MSORT; p += 1024)
        sk[p] = (p < NBOX) ? wsKey[p] : 0xFFFFFFFFFFFFFFFFull;
    __syncthreads();
    for (int k = 2; k <= MSORT; k <<= 1) {
        for (int j = k >> 1; j > 0; j >>= 1) {
            for (int i = tid; i < MSORT; i += 1024) {
                int ixj = i ^ j;
                if (ixj > i) {
                    unsigned long long a = sk[i], b = sk[ixj];
                    bool up = ((i & k) == 0);
                    if ((a > b) == up) { sk[i] = b; sk[ixj] = a; }
                }
            }
            __syncthreads();
        }
    }
    for (int p = tid; p < NBOX; p += 1024) {
        unsigned long long key = sk[p];
        unsigned idx = (unsigned)key;
        wsOrder[p] = idx;
        sBox[p]  = wsBox[idx];
        sArea[p] = wsArea[idx];
    }
    for (int p = tid; p < MSORT; p += 1024) {
        bool v = ((unsigned)(sk[p] >> 32)) < 0xFF800000u;   // score > -inf  <=>  valid
        if (p == 0 && !v) *wsV = 0u;
        if (p < MSORT - 1) {
            bool vn = ((unsigned)(sk[p+1] >> 32)) < 0xFF800000u;
            if (v && !vn) *wsV = (unsigned)(p + 1);
        }
    }
}

// ------------------------------------------------------------------
// Kernel 3: suppression mask.  One wave per 16x16 tile.
// WMMA computes union base area_i + area_j = [area,1] x [1;area].
// ------------------------------------------------------------------
__global__ __launch_bounds__(256)
void k_mask(const float4* __restrict__ sBox, const float* __restrict__ sArea,
            const unsigned* __restrict__ wsV, unsigned short* __restrict__ mask)
{
    const unsigned V = *wsV;
    const int lane = threadIdx.x & 31;
    const int wid  = blockIdx.x * 8 + (threadIdx.x >> 5);
    const int rowT = wid / NTILE, colT = wid % NTILE;   // rowT>=NTILE => rowBase>=8112>=V
    const int rowBase = rowT * TILE, colBase = colT * TILE;
    if ((unsigned)rowBase >= V || (unsigned)colBase >= V) return;

    // A (16x4): lanes 0-15 hold K=0,1; lanes 16-31 hold K=2,3 (zero rows)
    // B (4x16): row K=0 = ones, row K=1 = area_col, K=2,3 zero
    float aCol = sArea[colBase + (lane & 15)];
    v2f A, B;
    A.x = (lane < 16) ? sArea[rowBase + lane] : 0.f;
    A.y = (lane < 16) ? 1.f : 0.f;
    B.x = (lane < 16) ? 1.f : 0.f;
    B.y = (lane < 16) ? aCol : 0.f;
    v8f acc = {};
    acc = __builtin_amdgcn_wmma_f32_16x16x4_f32(false, A, false, B,
                                                (short)0, acc, false, false);
    // D layout: vgpr r, lane l -> M = r + 8*(l/16), N = l%16
    float4 cb = sBox[colBase + (lane & 15)];
    int rOff = (lane >> 4) << 3;
    for (int r = 0; r < 8; ++r) {
        float4 rb = sBox[rowBase + r + rOff];
        float ltx = fmaxf(rb.x, cb.x), lty = fmaxf(rb.y, cb.y);
        float rbx = fminf(rb.z, cb.z), rby = fminf(rb.w, cb.w);
        float iw = fmaxf(rbx - ltx, 0.f), ih = fmaxf(rby - lty, 0.f);
        float inter = iw * ih;
        float uni = acc[r] - inter;                 // area_M + area_N - inter via WMMA
        float iou = inter / fmaxf(uni, 1e-9f);
        bool sup = iou > IOU_T;
        unsigned bal = __builtin_amdgcn_ballot_w32(sup);
        if (lane == 0) {
            mask[(size_t)(rowBase + r)     * MASKW + colT] = (unsigned short)(bal & 0xFFFFu);
            mask[(size_t)(rowBase + r + 8) * MASKW + colT] = (unsigned short)(bal >> 16);
        }
    }
}

// ------------------------------------------------------------------
// Kernel 4: sequential greedy scan over the bitmask; scatter keep.
// ------------------------------------------------------------------
__global__ __launch_bounds__(256)
void k_scan(const unsigned short* __restrict__ mask, const unsigned* __restrict__ wsOrder,
            const unsigned* __restrict__ wsV, float* __restrict__ out)
{
    __shared__ unsigned short rem[MASKW];
    const int tid = threadIdx.x;
    const unsigned V = *wsV;
    for (int t = tid; t < MASKW; t += 256) rem[t] = 0;
    __syncthreads();
    for (unsigned i = 0; i < V; ++i) {
        bool kept = ((rem[i >> 4] >> (i & 15)) & 1) == 0;   // uniform across block
        __syncthreads();
        if (kept) {
            int t0 = (int)(i >> 4);
            const unsigned short* mrow = mask + (size_t)i * MASKW;
            for (int t = t0 + tid; t < NTILE; t += 256) {
                unsigned m = mrow[t];
                if (t == t0) m &= ~((1u << ((i & 15) + 1)) - 1u);  // only j > i suppressed
                rem[t] = (unsigned short)(rem[t] | m);
            }
        }
        __syncthreads();
    }
    for (int p = tid; p < NBOX; p += 256) {
        bool k = ((unsigned)p < V) && (((rem[p >> 4] >> (p & 15)) & 1) == 0);
        out[7 * NBOX + wsOrder[p]] = k ? 1.0f : 0.0f;
    }
}

// ------------------------------------------------------------------
extern "C" void kernel_launch(void* const* d_in, const int* in_sizes, int n_in,
                              void* d_out, int out_size, void* d_ws, size_t ws_size,
                              hipStream_t stream)
{
    const float* feat    = (const float*)d_in[0];   // 255*52*52
    const float* anchors = (const float*)d_in[1];   // 3*2
    float* out = (float*)d_out;                     // 7*8112 (stride*out) + 8112 (keep)
    char*  ws  = (char*)d_ws;

    float4* wsBox   = (float4*)(ws + WS_BOXES);
    float*  wsArea  = (float*)(ws + WS_AREA);
    unsigned long long* wsKey = (unsigned long long*)(ws + WS_KEY);
    unsigned* wsOrder = (unsigned*)(ws + WS_ORDER);
    float4* sBox    = (float4*)(ws + WS_SBOX);
    float*  sArea   = (float*)(ws + WS_SAREA);
    unsigned* wsV   = (unsigned*)(ws + WS_VCNT);
    unsigned short* mask = (unsigned short*)(ws + WS_MASK);

    k_decode<<<SDIM, 256, 0, stream>>>(feat, anchors, out, wsBox, wsArea, wsKey);
    k_sort<<<1, 1024, 0, stream>>>(wsKey, wsBox, wsArea, wsOrder, sBox, sArea, wsV);
    int nBlocks = (NTILE * NTILE + 7) / 8;          // 8 waves (tiles) per block
    k_mask<<<nBlocks, 256, 0, stream>>>(sBox, sArea, wsV, mask);
    k_scan<<<1, 256, 0, stream>>>(mask, wsOrder, wsV, out);
}